// lut_quant_fc_49538152792192
// MI455X (gfx1250) — compile-verified
//
#include <hip/hip_runtime.h>

typedef __attribute__((ext_vector_type(2))) float v2f;
typedef __attribute__((ext_vector_type(8))) float v8f;

#define IN_NUM   768
#define OUT_NUM  256
#define L0_      128
#define L1_      22
#define L2_      4
#define BR_      32   // B*R = 8*4 = wave32

// d_ws layout (float offsets)
#define N_WQ0  (256 * 128 * 64)          // 2,097,152  tanh-quantized w0, layout [o][l][e]
#define N_WQ1  (256 * 22 * 64)           //   360,448  [o][l][e]
#define N_WQ2  (256 * 4 * 64)            //    65,536
#define N_WQ3  (256 * 1 * 64)            //    16,384
#define N_P    (128 * 32 * 64)           //   262,144  p[l][br][e]
#define N_X1   (256 * 32 * 128)          // 1,048,576  x1[(o*32+br)*128 + l]
#define OFF_WQ0 0
#define OFF_WQ1 (OFF_WQ0 + N_WQ0)
#define OFF_WQ2 (OFF_WQ1 + N_WQ1)
#define OFF_WQ3 (OFF_WQ2 + N_WQ2)
#define OFF_P   (OFF_WQ3 + N_WQ3)
#define OFF_X1  (OFF_P + N_P)

// ---------------------------------------------------------------------------
// Prep: w_q = (tanh(w)+1)*0.5, elementwise (coalesced both sides)
// ---------------------------------------------------------------------------
__global__ void k_tanh_q(const float* __restrict__ src, float* __restrict__ dst, int n) {
    int i = blockIdx.x * blockDim.x + threadIdx.x;
    if (i < n) dst[i] = (tanhf(src[i]) + 1.0f) * 0.5f;
}

// ---------------------------------------------------------------------------
// Prep: soft one-hot p[l][br][e] = prod_level (bit if (e>>level)&1 else 1-bit)
// bit_level = x[br*768 + l*6 + level]. One wave per l, lane = br.
// ---------------------------------------------------------------------------
__global__ void k_build_p(const float* __restrict__ x, float* __restrict__ p) {
    int l  = blockIdx.x;      // 0..127
    int br = threadIdx.x;     // 0..31
    float bits[6];
#pragma unroll
    for (int t = 0; t < 6; ++t) bits[t] = x[br * IN_NUM + l * 6 + t];

    float acc[64];
    acc[0] = 1.0f;
#pragma unroll
    for (int t = 0; t < 6; ++t) {
        const int size = 1 << t;
        const float b = bits[t];
        const float nb = 1.0f - b;
#pragma unroll
        for (int j = 0; j < size; ++j) {
            acc[j + size] = acc[j] * b;   // bit t of index set
            acc[j]        = acc[j] * nb;  // bit t clear
        }
    }
    float* row = p + (l * 32 + br) * 64;
#pragma unroll
    for (int e = 0; e < 64; ++e) row[e] = acc[e];
}

// ---------------------------------------------------------------------------
// Layer 0 as 128 independent [32x64]x[64x256] f32 GEMMs via V_WMMA_F32_16X16X4_F32.
// Block = one l (128 blocks x 256 threads = 8 waves). Wave w does N-tiles {2w,2w+1}.
// A frag (16x4 f32): lane (k/2)*16+m holds A[m][k] in vgpr k&1  (per ISA layout).
// B frag mirrored: lane half selects K pair, lane&15 = column n.
// C/D (16x16 f32): vgpr r, lanes 0-15: M=r,N=lane; lanes 16-31: M=r+8,N=lane-16.
// ---------------------------------------------------------------------------
__global__ void k_layer0_wmma(const float* __restrict__ p,
                              const float* __restrict__ wq0,
                              float* __restrict__ x1) {
    const int l    = blockIdx.x;            // 0..127
    const int lane = threadIdx.x & 31;
    const int wave = threadIdx.x >> 5;      // 0..7
    const int h    = lane >> 4;             // lane half: selects K pair
    const int m    = lane & 15;

    const float* pbase = p + (l * 32) * 64; // p[l][br][e], e contiguous
    const float* arow0 = pbase + m * 64;        // br = m      (M-tile 0)
    const float* arow1 = pbase + (m + 16) * 64; // br = m + 16 (M-tile 1)

#pragma unroll
    for (int t = 0; t < 2; ++t) {
        const int n0 = (wave * 2 + t) * 16;
        const int o  = n0 + m;
        const float* brow = wq0 + (o * 128 + l) * 64;  // wq0[o][l][e], e contiguous

        v8f acc0 = {};   // br 0..15
        v8f acc1 = {};   // br 16..31
#pragma unroll
        for (int kk = 0; kk < 16; ++kk) {
            const int kb = kk * 4 + h * 2;               // e = kb, kb+1 for this half
            v2f a0 = *(const v2f*)(arow0 + kb);
            v2f a1 = *(const v2f*)(arow1 + kb);
            v2f b  = *(const v2f*)(brow + kb);
            acc0 = __builtin_amdgcn_wmma_f32_16x16x4_f32(false, a0, false, b,
                                                         (short)0, acc0, false, false);
            acc1 = __builtin_amdgcn_wmma_f32_16x16x4_f32(false, a1, false, b,
                                                         (short)0, acc1, false, false);
        }
        // Scatter D into x1[(o*32+br)*128 + l]
#pragma unroll
        for (int r = 0; r < 8; ++r) {
            const int br = r + 8 * h;
            x1[(o * 32 + br) * 128 + l]        = acc0[r];
            x1[(o * 32 + (br + 16)) * 128 + l] = acc1[r];
        }
    }
}

// ---------------------------------------------------------------------------
// 6-level LUT mux tree: 63 lerps, matches reference (1-bit)*y0 + bit*y1.
// w points at 64 wave-uniform weights in global memory (broadcast loads).
// ---------------------------------------------------------------------------
__device__ __forceinline__ float lut_tree(const float* __restrict__ w, const float bits[6]) {
    float t[32];
    {
        const float b = bits[0], nb = 1.0f - b;
#pragma unroll
        for (int j = 0; j < 32; ++j) t[j] = nb * w[2 * j] + b * w[2 * j + 1];
    }
#pragma unroll
    for (int lev = 1; lev < 6; ++lev) {
        const int sz = 32 >> lev;
        const float b = bits[lev], nb = 1.0f - b;
#pragma unroll
        for (int j = 0; j < sz; ++j) t[j] = nb * t[2 * j] + b * t[2 * j + 1];
    }
    return t[0];
}

// ---------------------------------------------------------------------------
// Layers 1..3 fused. One wave per o (lane = br). 32 blocks x 256 threads.
// Layer1: 22 cells over x1 row (pad 128..131 -> 0); layer2: 4 cells (pad 22,23);
// layer3: 1 cell (pad 4,5). Output [B,R,O] flat = br*256 + o.
// ---------------------------------------------------------------------------
__global__ void k_layers123(const float* __restrict__ x1,
                            const float* __restrict__ wq1,
                            const float* __restrict__ wq2,
                            const float* __restrict__ wq3,
                            float* __restrict__ out) {
    const int br = threadIdx.x & 31;
    const int o  = blockIdx.x * 8 + (threadIdx.x >> 5);
    const float* row = x1 + (o * 32 + br) * 128;

    float y1[24];
#pragma unroll
    for (int l = 0; l < 22; ++l) {
        float bits[6];
#pragma unroll
        for (int t = 0; t < 6; ++t) {
            const int idx = l * 6 + t;               // compile-time after unroll
            bits[t] = (idx < 128) ? row[idx] : 0.0f;
        }
        y1[l] = lut_tree(wq1 + (o * 22 + l) * 64, bits);
    }
    y1[22] = 0.0f;
    y1[23] = 0.0f;

    float y2[6];
#pragma unroll
    for (int l = 0; l < 4; ++l) {
        float bits[6];
#pragma unroll
        for (int t = 0; t < 6; ++t) bits[t] = y1[l * 6 + t];
        y2[l] = lut_tree(wq2 + (o * 4 + l) * 64, bits);
    }
    y2[4] = 0.0f;
    y2[5] = 0.0f;

    out[br * 256 + o] = lut_tree(wq3 + o * 64, y2);
}

// ---------------------------------------------------------------------------
extern "C" void kernel_launch(void* const* d_in, const int* in_sizes, int n_in,
                              void* d_out, int out_size, void* d_ws, size_t ws_size,
                              hipStream_t stream) {
    const float* x  = (const float*)d_in[0];  // [8,4,768]
    const float* w0 = (const float*)d_in[1];  // [256,128,64]
    const float* w1 = (const float*)d_in[2];  // [256,22,64]
    const float* w2 = (const float*)d_in[3];  // [256,4,64]
    const float* w3 = (const float*)d_in[4];  // [256,1,64]

    float* ws  = (float*)d_ws;
    float* wq0 = ws + OFF_WQ0;
    float* wq1 = ws + OFF_WQ1;
    float* wq2 = ws + OFF_WQ2;
    float* wq3 = ws + OFF_WQ3;
    float* p   = ws + OFF_P;
    float* x1  = ws + OFF_X1;

    k_tanh_q<<<(N_WQ0 + 255) / 256, 256, 0, stream>>>(w0, wq0, N_WQ0);
    k_tanh_q<<<(N_WQ1 + 255) / 256, 256, 0, stream>>>(w1, wq1, N_WQ1);
    k_tanh_q<<<(N_WQ2 + 255) / 256, 256, 0, stream>>>(w2, wq2, N_WQ2);
    k_tanh_q<<<(N_WQ3 + 255) / 256, 256, 0, stream>>>(w3, wq3, N_WQ3);
    k_build_p<<<128, 32, 0, stream>>>(x, p);
    k_layer0_wmma<<<128, 256, 0, stream>>>(p, wq0, x1);
    k_layers123<<<32, 256, 0, stream>>>(x1, wq1, wq2, wq3, (float*)d_out);
}